// RickerPredation_46961172414724
// MI455X (gfx1250) — compile-verified
//
#include <hip/hip_runtime.h>
#include <hip/hip_bf16.h>
#include <stdint.h>

// ---------------------------------------------------------------------------
// Ricker predation scan, T = 65536.
// Strictly serial recurrence => minimize per-step dependent-op count.
//   n1 = s1 * exp2(E1[t] - C11*s1 - C12*s2)     (log2e folded into E/C)
//   n2 = s2 * exp2(E2[t] - C22*s2 - C21*s1)
// E1/E2 precomputed fully in parallel (kernel A). Serial kernel (B) is one
// wave: lane 0 runs the chain, the wave double-buffers (E1,E2) chunks into
// LDS with gfx1250 async global->LDS loads (ASYNCcnt-tracked).
// ---------------------------------------------------------------------------

#define RK_CH 128   // steps per LDS chunk (1 KB of float2)

__global__ __launch_bounds__(256)
void ricker_pre(const float* __restrict__ f, const float* __restrict__ p,
                float2* __restrict__ ew, int T) {
  int t = blockIdx.x * blockDim.x + threadIdx.x;
  if (t >= T) return;
  const float L2E = 1.4426950408889634f;
  const float k1 = L2E * p[0];            // log2e * alpha1
  const float k2 = L2E * p[5];            // log2e * alpha2
  float2 r;
  if (t == 0) {
    // Synthetic E[0] so stepping the pre-state (1,1) yields exactly the
    // initial condition: E = C11 + C12 (built from the same products the
    // scan kernel uses, so cancellation is near-exact).
    r.x = k1 * p[1] + k1 * p[2];
    r.y = k2 * p[6] + k2 * p[7];
  } else {
    float x  = f[t - 1];
    float x2 = x * x;
    // k * (1 + bx*x + cx*x^2)
    r.x = k1 * (1.0f + __builtin_fmaf(p[4], x2, p[3] * x));
    r.y = k2 * (1.0f + __builtin_fmaf(p[9], x2, p[8] * x));
  }
  ew[t] = r;
}

__device__ __forceinline__
void rk_issue_chunk(const float2* ew, int c, uint32_t lds_base, int tid) {
  // Copy 1 KB (RK_CH float2) with two async b128 loads per lane.
  // INST_OFFSET is added to BOTH the LDS and global addresses (ISA 08 §4.4).
  unsigned long long g = (unsigned long long)(const void*)(ew + (size_t)c * RK_CH)
                       + (unsigned long long)(tid * 16);
  uint32_t l = lds_base + (uint32_t)(tid * 16);
  asm volatile("global_load_async_to_lds_b128 %0, %1, off"
               :: "v"(l), "v"(g) : "memory");
  asm volatile("global_load_async_to_lds_b128 %0, %1, off offset:512"
               :: "v"(l), "v"(g) : "memory");
}

__global__ __launch_bounds__(32)
void ricker_scan(const float2* __restrict__ ew, const float* __restrict__ p,
                 float* __restrict__ out, int T) {
  __shared__ __align__(16) float2 lbuf[2][RK_CH];
  const int tid = threadIdx.x;
  const float L2E = 1.4426950408889634f;
  // Negated fused coefficients (negation of a product is exact).
  const float nC11 = -L2E * p[0] * p[1];
  const float nC12 = -L2E * p[0] * p[2];
  const float nC22 = -L2E * p[5] * p[6];
  const float nC21 = -L2E * p[5] * p[7];

  const int NCH = T / RK_CH;
  const uint32_t lds0 = (uint32_t)(uintptr_t)(&lbuf[0][0]);
  const uint32_t lds1 = (uint32_t)(uintptr_t)(&lbuf[1][0]);

  rk_issue_chunk(ew, 0, lds0, tid);
  rk_issue_chunk(ew, 1, lds1, tid);

  float s1 = 1.0f, s2 = 1.0f;
  for (int c = 0; c < NCH; ++c) {
    // Chunk c complete when <=2 async ops (chunk c+1's) remain outstanding.
    if (c + 1 < NCH) asm volatile("s_wait_asynccnt 0x2" ::: "memory");
    else             asm volatile("s_wait_asynccnt 0x0" ::: "memory");

    if (tid == 0) {
      const float2* e  = lbuf[c & 1];
      float* o1 = out + (size_t)c * RK_CH;
      float* o2 = o1 + T;
      #pragma unroll 2
      for (int i = 0; i < RK_CH; i += 4) {
        float b1v[4], b2v[4];
        #pragma unroll
        for (int j = 0; j < 4; ++j) {
          float2 ev = e[i + j];
          // Critical path per species: fma -> fma -> exp2 -> mul.
          float x1 = __builtin_fmaf(nC12, s2, __builtin_fmaf(nC11, s1, ev.x));
          float x2 = __builtin_fmaf(nC21, s1, __builtin_fmaf(nC22, s2, ev.y));
          s1 = s1 * __builtin_amdgcn_exp2f(x1);
          s2 = s2 * __builtin_amdgcn_exp2f(x2);
          b1v[j] = s1;
          b2v[j] = s2;
        }
        *(float4*)(o1 + i) = make_float4(b1v[0], b1v[1], b1v[2], b1v[3]);
        *(float4*)(o2 + i) = make_float4(b2v[0], b2v[1], b2v[2], b2v[3]);
      }
    }
    // Lane 0 finished reading buf[c&1]; safe to overwrite it with chunk c+2.
    asm volatile("s_wait_dscnt 0x0" ::: "memory");
    if (c + 2 < NCH) rk_issue_chunk(ew, c + 2, (c & 1) ? lds1 : lds0, tid);
  }
}

// Fallback: plain single-lane serial scan (no workspace needed).
__global__ __launch_bounds__(32)
void ricker_serial(const float* __restrict__ f, const float* __restrict__ p,
                   float* __restrict__ out, int T) {
  if (threadIdx.x != 0) return;
  const float L2E = 1.4426950408889634f;
  const float k1 = L2E * p[0], k2 = L2E * p[5];
  const float nC11 = -k1 * p[1], nC12 = -k1 * p[2];
  const float nC22 = -k2 * p[6], nC21 = -k2 * p[7];
  const float bx1 = p[3], cx1 = p[4], bx2 = p[8], cx2 = p[9];
  float s1 = 1.0f, s2 = 1.0f;
  out[0] = 1.0f; out[T] = 1.0f;
  for (int t = 1; t < T; ++t) {
    float x  = f[t - 1];
    float xx = x * x;
    float e1 = k1 * (1.0f + __builtin_fmaf(cx1, xx, bx1 * x));
    float e2 = k2 * (1.0f + __builtin_fmaf(cx2, xx, bx2 * x));
    float x1 = __builtin_fmaf(nC12, s2, __builtin_fmaf(nC11, s1, e1));
    float x2 = __builtin_fmaf(nC21, s1, __builtin_fmaf(nC22, s2, e2));
    s1 = s1 * __builtin_amdgcn_exp2f(x1);
    s2 = s2 * __builtin_amdgcn_exp2f(x2);
    out[t] = s1; out[T + t] = s2;
  }
}

extern "C" void kernel_launch(void* const* d_in, const int* in_sizes, int n_in,
                              void* d_out, int out_size, void* d_ws, size_t ws_size,
                              hipStream_t stream) {
  const float* forcing = (const float*)d_in[0];
  const float* params  = (const float*)d_in[1];
  float* out = (float*)d_out;
  const int T = in_sizes[0];   // 65536

  const bool fast = (ws_size >= (size_t)T * sizeof(float2)) &&
                    (T % RK_CH == 0) && (T / RK_CH >= 2);
  if (fast) {
    float2* ew = (float2*)d_ws;
    ricker_pre<<<(T + 255) / 256, 256, 0, stream>>>(forcing, params, ew, T);
    ricker_scan<<<1, 32, 0, stream>>>(ew, params, out, T);
  } else {
    ricker_serial<<<1, 32, 0, stream>>>(forcing, params, out, T);
  }
}